// LSTMCell_66554813218861
// MI455X (gfx1250) — compile-verified
//
#include <hip/hip_runtime.h>

// LSTM cell, D=8192. Bandwidth-bound GEMV x4 via V_WMMA_F32_16X16X4_F32.
// Fused: gate GEMVs + bias + sigmoid/tanh combine in one kernel.

typedef __attribute__((ext_vector_type(2))) float v2f;
typedef __attribute__((ext_vector_type(4))) float v4f;
typedef __attribute__((ext_vector_type(8))) float v8f;

#define LSTM_D    8192
#define LSTM_2D   16384

__device__ __forceinline__ float sigmoidf_fast(float v) {
    return 1.0f / (1.0f + __expf(-v));
}

__global__ __launch_bounds__(256) void lstm_cell_wmma(
    const float* __restrict__ x,  const float* __restrict__ h,
    const float* __restrict__ c,
    const float* __restrict__ Wf, const float* __restrict__ bf,
    const float* __restrict__ Wi, const float* __restrict__ bi,
    const float* __restrict__ Wc, const float* __restrict__ bc,
    const float* __restrict__ Wo, const float* __restrict__ bo,
    float* __restrict__ out)
{
    __shared__ float sConcat[LSTM_2D];   // 64 KB: [h(8192) | x(8192)]
    __shared__ float sGates[4][32];      // pre-activations for this block's 32 cols

    const int tid = threadIdx.x;

    // Stage concat into LDS: 16384 floats, 256 threads, float4 moves.
    for (int i = tid * 4; i < LSTM_D; i += 256 * 4) {
        *(v4f*)&sConcat[i]          = *(const v4f*)&h[i];
        *(v4f*)&sConcat[LSTM_D + i] = *(const v4f*)&x[i];
    }
    __syncthreads();

    const int wave    = tid >> 5;     // 8 waves (wave32)
    const int lane    = tid & 31;
    const int gate    = wave & 3;     // 0=f 1=i 2=c 3=o
    const int tileIdx = wave >> 2;    // 0 or 1
    const int n0      = blockIdx.x * 32 + tileIdx * 16;
    const int half    = lane >> 4;    // lane group: 0 -> K {0,1}, 1 -> K {2,3}
    const int l       = lane & 15;    // column within 16-wide tile

    const float* W = (gate == 0) ? Wf : (gate == 1) ? Wi : (gate == 2) ? Wc : Wo;
    const float* B = (gate == 0) ? bf : (gate == 1) ? bi : (gate == 2) ? bc : bo;

    // Weight pointer for this lane: row (k + 2*half), column n0+l.
    const float* wp = W + (size_t)(2 * half) * LSTM_D + (size_t)n0 + (size_t)l;
    // A operand from LDS: concat[k + 2*half], concat[k + 2*half + 1].
    const float* ap = sConcat + 2 * half;

    v8f acc0 = {0.f, 0.f, 0.f, 0.f, 0.f, 0.f, 0.f, 0.f};
    v8f acc1 = {0.f, 0.f, 0.f, 0.f, 0.f, 0.f, 0.f, 0.f};

    // K loop: 16384 total, 8 K per iteration (two 16x16x4 WMMAs, two acc chains).
#pragma unroll 4
    for (int k = 0; k < LSTM_2D; k += 8) {
        v2f a0 = *(const v2f*)(ap + k);
        v2f a1 = *(const v2f*)(ap + k + 4);
        v2f b0, b1;
        b0.x = wp[(size_t)(k + 0) * LSTM_D];
        b0.y = wp[(size_t)(k + 1) * LSTM_D];
        b1.x = wp[(size_t)(k + 4) * LSTM_D];
        b1.y = wp[(size_t)(k + 5) * LSTM_D];
        // 8 args: (neg_a, A, neg_b, B, c_mod, C, reuse_a, reuse_b)
        acc0 = __builtin_amdgcn_wmma_f32_16x16x4_f32(false, a0, false, b0,
                                                     (short)0, acc0, false, false);
        acc1 = __builtin_amdgcn_wmma_f32_16x16x4_f32(false, a1, false, b1,
                                                     (short)0, acc1, false, false);
    }

    // Every row of D equals the GEMV result (A rows replicated).
    // Row 0 lives in acc element 0 of lanes 0..15.
    if (half == 0) {
        float pre = acc0[0] + acc1[0] + B[n0 + l];
        sGates[gate][tileIdx * 16 + l] = pre;
    }
    __syncthreads();

    // Elementwise LSTM combine for this block's 32 columns.
    if (tid < 32) {
        const int n = blockIdx.x * 32 + tid;
        const float zf = sigmoidf_fast(sGates[0][tid]);
        const float zi = sigmoidf_fast(sGates[1][tid]);
        const float z  = tanhf(sGates[2][tid]);
        const float zo = sigmoidf_fast(sGates[3][tid]);
        const float c_new = c[n] * zf + z * zi;
        out[n] = zo * tanhf(c_new);
    }
}

extern "C" void kernel_launch(void* const* d_in, const int* in_sizes, int n_in,
                              void* d_out, int out_size, void* d_ws, size_t ws_size,
                              hipStream_t stream) {
    const float* x  = (const float*)d_in[0];
    const float* h  = (const float*)d_in[1];
    const float* c  = (const float*)d_in[2];
    const float* Wf = (const float*)d_in[3];
    const float* bf = (const float*)d_in[4];
    const float* Wi = (const float*)d_in[5];
    const float* bi = (const float*)d_in[6];
    const float* Wc = (const float*)d_in[7];
    const float* bc = (const float*)d_in[8];
    const float* Wo = (const float*)d_in[9];
    const float* bo = (const float*)d_in[10];
    float* out = (float*)d_out;

    // 256 blocks x 32 columns = 8192 outputs; each block fuses all 4 gates.
    lstm_cell_wmma<<<LSTM_D / 32, 256, 0, stream>>>(
        x, h, c, Wf, bf, Wi, bi, Wc, bc, Wo, bo, out);
}